// QGraphConv_1864015807109
// MI455X (gfx1250) — compile-verified
//
#include <hip/hip_runtime.h>

// ---- problem constants (from reference) ----
#define N_NODES   50000
#define N_EDGES   800000
#define IN_FEATS  256
#define OUT_FEATS 64

typedef __attribute__((ext_vector_type(2))) float v2f;
typedef __attribute__((ext_vector_type(8))) float v8f;

// ------------------------------------------------------------------
// Kernel 1: degree histograms (float atomics so no int->float pass)
// ------------------------------------------------------------------
__global__ void gcn_degrees(const int* __restrict__ src,
                            const int* __restrict__ dst,
                            float* __restrict__ outdeg,
                            float* __restrict__ indeg,
                            int nEdges)
{
    int e = blockIdx.x * blockDim.x + threadIdx.x;
    if (e >= nEdges) return;
    unsafeAtomicAdd(&outdeg[src[e]], 1.0f);
    unsafeAtomicAdd(&indeg[dst[e]], 1.0f);
}

// ------------------------------------------------------------------
// Kernel 2: h = (feat * rsqrt(outdeg)) @ W^T  via V_WMMA_F32_16X16X4_F32
// One wave computes a full 16x64 strip of h (4 column tiles), so each
// A fetch feeds 4 WMMAs: feat is streamed exactly once (51.2 MB).
// A (16x4 f32):  lane m=lane&15 holds rows; lanes 0-15 -> K=k0+0..1,
//                lanes 16-31 -> K=k0+2..3 (2 VGPRs).
// B (4x16 f32):  mirrored: lane n=lane&15 holds cols, same K split.
// C/D (16x16):   VGPR r -> M = r + 8*(lane>=16), N = lane&15.
// ------------------------------------------------------------------
__global__ void gcn_gemm_wmma(const float* __restrict__ feat,
                              const float* __restrict__ weight,
                              const float* __restrict__ outdeg,
                              float* __restrict__ h)
{
    const int lane = threadIdx.x & 31;
    const int wave = threadIdx.x >> 5;
    const int rowTile = blockIdx.x * 8 + wave;   // 0 .. 3124
    if (rowTile >= (N_NODES / 16)) return;       // wave-uniform: EXEC stays all-1s

    const int m       = lane & 15;               // row (A) / col (B) within tile
    const int kOff    = (lane >> 4) << 1;        // 0 for lanes 0-15, 2 for lanes 16-31
    const int rowBase = rowTile * 16;

    // left norm: rsqrt(clip(out_deg,1)) for this lane's A row
    const float s = __frsqrt_rn(fmaxf(outdeg[rowBase + m], 1.0f));

    const float* __restrict__ aPtr = feat   + (size_t)(rowBase + m) * IN_FEATS + kOff;
    const float* __restrict__ bPtr = weight + (size_t)m * IN_FEATS + kOff;   // col tile t at +t*16*IN_FEATS

    v8f c0 = {}, c1 = {}, c2 = {}, c3 = {};
    #pragma unroll 4
    for (int k0 = 0; k0 < IN_FEATS; k0 += 4) {
        v2f a = *(const v2f*)(aPtr + k0);
        a *= s;                                  // fold left-norm into A
        v2f b0 = *(const v2f*)(bPtr + k0);
        v2f b1 = *(const v2f*)(bPtr + 16 * IN_FEATS + k0);
        v2f b2 = *(const v2f*)(bPtr + 32 * IN_FEATS + k0);
        v2f b3 = *(const v2f*)(bPtr + 48 * IN_FEATS + k0);
        c0 = __builtin_amdgcn_wmma_f32_16x16x4_f32(false, a, false, b0, (short)0, c0, false, false);
        c1 = __builtin_amdgcn_wmma_f32_16x16x4_f32(false, a, false, b1, (short)0, c1, false, false);
        c2 = __builtin_amdgcn_wmma_f32_16x16x4_f32(false, a, false, b2, (short)0, c2, false, false);
        c3 = __builtin_amdgcn_wmma_f32_16x16x4_f32(false, a, false, b3, (short)0, c3, false, false);
    }

    // store D: VGPR r -> row (r + 8*(lane>=16)), col tile t -> cols t*16 + m
    const int mStore = (lane >> 4) << 3;          // 0 or 8
    float* __restrict__ hOut = h + (size_t)(rowBase + mStore) * OUT_FEATS + m;
    #pragma unroll
    for (int r = 0; r < 8; ++r) {
        hOut[(size_t)r * OUT_FEATS +  0] = c0[r];
        hOut[(size_t)r * OUT_FEATS + 16] = c1[r];
        hOut[(size_t)r * OUT_FEATS + 32] = c2[r];
        hOut[(size_t)r * OUT_FEATS + 48] = c3[r];
    }
}

// ------------------------------------------------------------------
// Kernel 3: edge scatter-sum: out[dst] += h[src]. 16 threads/edge,
// float4 gather (L2-resident: h is 12.8MB << 192MB) + 4 native fp32 atomics.
// ------------------------------------------------------------------
__global__ void gcn_scatter(const float* __restrict__ h,
                            const int* __restrict__ src,
                            const int* __restrict__ dst,
                            float* __restrict__ out,
                            int nEdges)
{
    int tid = blockIdx.x * blockDim.x + threadIdx.x;
    int e = tid >> 4;
    if (e >= nEdges) return;
    int q = (tid & 15) << 2;                  // feature chunk of 4
    int s = src[e];
    int d = dst[e];
    const float4 v = *(const float4*)(h + (size_t)s * OUT_FEATS + q);
    float* __restrict__ o = out + (size_t)d * OUT_FEATS + q;
    unsafeAtomicAdd(o + 0, v.x);
    unsafeAtomicAdd(o + 1, v.y);
    unsafeAtomicAdd(o + 2, v.z);
    unsafeAtomicAdd(o + 3, v.w);
}

// ------------------------------------------------------------------
// Kernel 4: right norm + bias (in place on d_out)
// ------------------------------------------------------------------
__global__ void gcn_finalize(float* __restrict__ out,
                             const float* __restrict__ indeg,
                             const float* __restrict__ bias,
                             int total)
{
    int tid = blockIdx.x * blockDim.x + threadIdx.x;
    if (tid >= total) return;
    int n = tid >> 6;                         // OUT_FEATS == 64
    int c = tid & 63;
    out[tid] = out[tid] * __frsqrt_rn(fmaxf(indeg[n], 1.0f)) + bias[c];
}

// ------------------------------------------------------------------
extern "C" void kernel_launch(void* const* d_in, const int* in_sizes, int n_in,
                              void* d_out, int out_size, void* d_ws, size_t ws_size,
                              hipStream_t stream)
{
    const float* feat   = (const float*)d_in[0];   // [N_NODES, IN_FEATS]
    const float* weight = (const float*)d_in[1];   // [OUT_FEATS, IN_FEATS]
    const float* bias   = (const float*)d_in[2];   // [OUT_FEATS]
    const int*   src    = (const int*)d_in[3];     // [N_EDGES]
    const int*   dst    = (const int*)d_in[4];     // [N_EDGES]
    float*       out    = (float*)d_out;           // [N_NODES, OUT_FEATS]

    // workspace layout
    float* outdeg = (float*)d_ws;                       // N_NODES
    float* indeg  = outdeg + N_NODES;                   // N_NODES
    float* h      = indeg + N_NODES;                    // N_NODES * OUT_FEATS

    // 0) zero accumulators (capture-safe stream memsets)
    hipMemsetAsync(outdeg, 0, (size_t)2 * N_NODES * sizeof(float), stream);
    hipMemsetAsync(out, 0, (size_t)out_size * sizeof(float), stream);

    // 1) degrees
    {
        int threads = 256;
        int blocks = (N_EDGES + threads - 1) / threads;
        gcn_degrees<<<blocks, threads, 0, stream>>>(src, dst, outdeg, indeg, N_EDGES);
    }

    // 2) WMMA GEMM: 3125 row-strip waves (16x64 out per wave), 8 waves/block
    {
        int totalWaves = N_NODES / 16;               // 3125
        int blocks = (totalWaves + 7) / 8;           // 391
        gcn_gemm_wmma<<<blocks, 256, 0, stream>>>(feat, weight, outdeg, h);
    }

    // 3) edge scatter: 16 threads per edge
    {
        long long totalThreads = (long long)N_EDGES * 16;
        int threads = 256;
        int blocks = (int)((totalThreads + threads - 1) / threads);
        gcn_scatter<<<blocks, threads, 0, stream>>>(h, src, dst, out, N_EDGES);
    }

    // 4) finalize
    {
        int total = N_NODES * OUT_FEATS;
        int threads = 256;
        int blocks = (total + threads - 1) / threads;
        gcn_finalize<<<blocks, threads, 0, stream>>>(out, indeg, bias, total);
    }
}